// SSMConv1D_66159676228093
// MI455X (gfx1250) — compile-verified
//
#include <hip/hip_runtime.h>
#include <hip/hip_bf16.h>
#include <stdint.h>

#define N_BATCH 32
#define T_LEN   4096
#define CH      512
#define KS      7
#define HALO    3
#define TT      128                 // t-rows (outputs) per block
#define CHUNK   8                   // rows per async chunk (16 KB)
#define NCHUNK  (TT / CHUNK)        // 16 compute chunks; NCHUNK+1 load chunks
#define RING    32                  // LDS ring rows (64 KB) -> slot = row & 31
#define ROW_B   (CH * 4)            // 2048 bytes per row
#define BLOCK   256                 // 8 waves (wave32)

typedef float f4 __attribute__((ext_vector_type(4)));

// CDNA5 async global->LDS copy (tracked by ASYNCcnt), non-temporal: x has zero
// reuse beyond tiny halos and is larger than the 192MB L2.
__device__ __forceinline__ void async_b128_to_lds(uint32_t lds_off,
                                                  unsigned long long gaddr) {
  asm volatile("global_load_async_to_lds_b128 %0, %1, off th:TH_LOAD_NT"
               :: "v"(lds_off), "v"(gaddr) : "memory");
}

__device__ __forceinline__ uint32_t ring_slot(int row) {
  return (uint32_t)(row + RING) & (RING - 1);  // row >= -3, consistent mod 32
}

template <bool EDGE>
__device__ __forceinline__ void conv_body(const float* __restrict__ x,
                                          float* __restrict__ out,
                                          float* lds, int n, int t0,
                                          int tid, const f4* w) {
  const int col = tid & 127;   // which float4 of the 512 channels
  const int par = tid >> 7;    // row parity: 2 rows in flight across the block

  const uint32_t lds_base =
      (uint32_t)(uintptr_t)lds;  // generic shared ptr low 32 bits = LDS offset
  const unsigned long long xbase =
      (unsigned long long)(uintptr_t)x +
      (unsigned long long)n * T_LEN * ROW_B;

  // Chunk c stages input rows [t0-3+8c, t0+5+8c) into the ring.
  // Always exactly 4 async b128 per wave so ASYNCcnt bookkeeping is uniform.
  auto issue_chunk = [&](int c) {
    const int rb = t0 - HALO + c * CHUNK;
#pragma unroll
    for (int i = 0; i < 4; ++i) {
      int off   = (i * BLOCK + tid) * 16;       // 0..16383 within the 16KB chunk
      int row   = rb + (off >> 11);
      int cbyte = off & (ROW_B - 1);
      int rowc  = row;
      if (EDGE)  // clamp OOB halo rows to a valid address; zeroed on read
        rowc = row < 0 ? 0 : (row > T_LEN - 1 ? T_LEN - 1 : row);
      uint32_t loff = lds_base + ring_slot(row) * ROW_B + (uint32_t)cbyte;
      unsigned long long g = xbase + (unsigned long long)rowc * ROW_B + cbyte;
      async_b128_to_lds(loff, g);
    }
  };

  const f4* lrow = (const f4*)lds;  // [slot * 128 + col]
  auto load_row = [&](int row) -> f4 {
    f4 v = lrow[ring_slot(row) * (CH / 4) + col];
    if (EDGE) {
      if (row < 0 || row >= T_LEN) v = (f4){0.f, 0.f, 0.f, 0.f};
    }
    return v;
  };

  issue_chunk(0);
  issue_chunk(1);
  issue_chunk(2);

  f4 win[KS];  // register sliding window, rows t-3..t+3 (stride-2 advance)

  for (int c = 0; c < NCHUNK; ++c) {
    // Prefetch 3 chunks ahead; async loads complete in order, so allowing
    // N outstanding means everything older than the newest N has landed.
    if (c + 3 <= NCHUNK) {
      issue_chunk(c + 3);
      asm volatile("s_wait_asynccnt 0x8" ::: "memory");  // chunks c+2,c+3 in flight
    } else if (c + 2 <= NCHUNK) {
      asm volatile("s_wait_asynccnt 0x4" ::: "memory");  // chunk c+2 in flight
    } else {
      asm volatile("s_wait_asynccnt 0x0" ::: "memory");
    }
    __syncthreads();  // all waves' chunks <= c+1 now resident in LDS

    const int tc = t0 + c * CHUNK;
#pragma unroll
    for (int i = 0; i < 4; ++i) {
      const int t = tc + par + 2 * i;
      if (c == 0 && i == 0) {
#pragma unroll
        for (int k = 0; k < KS; ++k) win[k] = load_row(t - HALO + k);
      } else {
#pragma unroll
        for (int k = 0; k < KS - 2; ++k) win[k] = win[k + 2];
        win[5] = load_row(t + 2);
        win[6] = load_row(t + 3);
      }
      f4 acc = w[0] * win[0];
#pragma unroll
      for (int k = 1; k < KS; ++k) acc += w[k] * win[k];
      f4* dst = (f4*)(out + ((size_t)n * T_LEN + t) * CH) + col;
      __builtin_nontemporal_store(acc, dst);  // streaming store, skip L2 rinse
    }
    __syncthreads();  // readers done before ring slots are overwritten
  }
}

__global__ __launch_bounds__(BLOCK)
void SSMConv1D_kernel(const float* __restrict__ x,
                      const float* __restrict__ A,
                      const float* __restrict__ B,
                      const float* __restrict__ C,
                      float* __restrict__ out) {
  __shared__ __align__(16) float lds[RING * CH];

  const int tid  = threadIdx.x;
  const int bx   = blockIdx.x;
  const int n    = bx / (T_LEN / TT);
  const int tile = bx % (T_LEN / TT);
  const int t0   = tile * TT;

  // Per-channel 7-tap kernel: w[k] = (B*C) * A^k  (4 channels per thread)
  const int col = tid & 127;
  f4 a  = ((const f4*)A)[col];
  f4 cb = ((const f4*)B)[col] * ((const f4*)C)[col];
  f4 w[KS];
  f4 p = {1.f, 1.f, 1.f, 1.f};
#pragma unroll
  for (int k = 0; k < KS; ++k) { w[k] = cb * p; p *= a; }

  // Only the first/last t-tile of each batch touches the zero halo; interior
  // tiles (15/16 of all blocks) run without clamps or OOB selects.
  if (tile == 0 || tile == (T_LEN / TT) - 1) {
    conv_body<true>(x, out, lds, n, t0, tid, w);
  } else {
    conv_body<false>(x, out, lds, n, t0, tid, w);
  }
}

extern "C" void kernel_launch(void* const* d_in, const int* in_sizes, int n_in,
                              void* d_out, int out_size, void* d_ws, size_t ws_size,
                              hipStream_t stream) {
  const float* x = (const float*)d_in[0];
  const float* A = (const float*)d_in[1];
  const float* B = (const float*)d_in[2];
  const float* C = (const float*)d_in[3];
  float* out = (float*)d_out;
  dim3 grid(N_BATCH * (T_LEN / TT));  // 1024 blocks
  SSMConv1D_kernel<<<grid, BLOCK, 0, stream>>>(x, A, B, C, out);
}